// Pointer_Generator_19851338842600
// MI455X (gfx1250) — compile-verified
//
#include <hip/hip_runtime.h>
#include <hip/hip_bf16.h>

#define B_   16
#define H_   12
#define T_   128
#define L_   512
#define D_   768
#define V_   32000

typedef __attribute__((ext_vector_type(2))) float v2f;
typedef __attribute__((ext_vector_type(8))) float v8f;

// ---------------------------------------------------------------------------
// Kernel 1: p_gen = sigmoid(dec_output @ W_pg + b_pg) via V_WMMA_F32_16X16X4_F32
// One wave (32 threads) computes 16 rows; K-loop over D=768 in steps of 4.
// A (16x4 f32) layout: lanes 0-15 hold K=k,k+1 ; lanes 16-31 hold K=k+2,k+3.
// B (4x16 f32) column N=0 only: lane 0 holds B[0,0],B[1,0]; lane 16 holds B[2,0],B[3,0].
// D (16x16 f32): column 0 -> lane 0 (rows 0-7 in v[0..7]), lane 16 (rows 8-15).
// ---------------------------------------------------------------------------
__global__ __launch_bounds__(32)
void pgen_wmma_kernel(const float* __restrict__ dec,   // [2048, 768]
                      const float* __restrict__ Wp,    // [768]
                      const float* __restrict__ bp,    // [1]
                      float* __restrict__ pgen)        // [2048]
{
    const int lane = threadIdx.x;        // 0..31
    const int r    = lane & 15;
    const int hi   = lane >> 4;          // 0 or 1
    const int rowBase = blockIdx.x * 16; // 128 blocks * 16 rows = 2048

    const float* rowPtr = dec + (size_t)(rowBase + r) * D_;
    const bool col0 = (r == 0);          // hoisted lane predicate -> v_cndmask

    v8f acc = {};
    for (int k = 0; k < D_; k += 4) {
        // A: each lane loads 2 consecutive f32 (8B aligned: k%4==0, +2*hi)
        float2 av = *(const float2*)(rowPtr + k + 2 * hi);
        v2f a; a[0] = av.x; a[1] = av.y;

        // B: ALL lanes load the weight pair (broadcast-coalesced b64 load),
        // then branch-free select so only lanes 0/16 carry column N=0.
        float2 wv = *(const float2*)(Wp + k + 2 * hi);
        v2f b;
        b[0] = col0 ? wv.x : 0.0f;
        b[1] = col0 ? wv.y : 0.0f;

        acc = __builtin_amdgcn_wmma_f32_16x16x4_f32(
            /*neg_a=*/false, a, /*neg_b=*/false, b,
            /*c_mod=*/(short)0, acc, /*reuse_a=*/false, /*reuse_b=*/false);
    }

    if (col0) {
        const float bias = bp[0];
        const int base = rowBase + hi * 8;
        #pragma unroll
        for (int i = 0; i < 8; ++i) {
            float z = acc[i] + bias;
            pgen[base + i] = 1.0f / (1.0f + __expf(-z));
        }
    }
}

// ---------------------------------------------------------------------------
// Kernel 2: fully fused per-(b,t) row:
//   - zero 128KB LDS scatter table (fits thanks to CDNA5 320KB WGP LDS)
//   - head-mean + softmax(512) of attention, scatter-add (1-pg)*attn via ds_add_f32
//   - online max/sum over the 32000-wide vocab row
//   - out = log(pg * softmax(final) + scatter)
// ---------------------------------------------------------------------------
#define BLK 1024

__global__ __launch_bounds__(BLK)
void fused_row_kernel(const float* __restrict__ fin,   // [2048, 32000]
                      const float* __restrict__ attn,  // [16,12,128,512]
                      const int*   __restrict__ enc,   // [16,512]
                      const float* __restrict__ pgen,  // [2048]
                      float* __restrict__ out)         // [2048, 32000]
{
    extern __shared__ float smem[];
    float* scat    = smem;               // V_ floats (128000 B)
    float* attnbuf = smem + V_;          // 512 floats
    float* red     = attnbuf + L_;       // 64 floats reduction scratch

    const int t   = blockIdx.x;
    const int b   = blockIdx.y;
    const int row = b * T_ + t;
    const int tid = threadIdx.x;
    const int lane = tid & 31;
    const int wave = tid >> 5;

    // 1) zero scatter table
    for (int v = tid; v < V_; v += BLK) scat[v] = 0.0f;

    // 2) mean over heads -> attnbuf
    if (tid < L_) {
        const float* ap = attn + ((size_t)(b * H_) * T_ + t) * L_ + tid;
        float s = 0.0f;
        #pragma unroll
        for (int h = 0; h < H_; ++h) s += ap[(size_t)h * T_ * L_];
        attnbuf[tid] = s * (1.0f / (float)H_);
    }
    __syncthreads();

    // 3) softmax stats over attnbuf (wave 0 only)
    if (tid < 32) {
        float m = -INFINITY;
        for (int j = lane; j < L_; j += 32) m = fmaxf(m, attnbuf[j]);
        #pragma unroll
        for (int off = 16; off > 0; off >>= 1)
            m = fmaxf(m, __shfl_xor(m, off));
        float s = 0.0f;
        for (int j = lane; j < L_; j += 32) s += __expf(attnbuf[j] - m);
        #pragma unroll
        for (int off = 16; off > 0; off >>= 1)
            s += __shfl_xor(s, off);
        if (lane == 0) { red[0] = m; red[1] = s; }
    }
    __syncthreads();

    const float M1 = red[0];
    const float S1 = red[1];
    const float pg = pgen[row];
    __syncthreads();   // everyone has read red[0..1] before it is reused below

    // 4) scatter-add (1-pg)*attn_dist into LDS (ds_add_f32 atomics)
    if (tid < L_) {
        float c = (1.0f - pg) * __expf(attnbuf[tid] - M1) / S1;
        int id = enc[b * L_ + tid];
        atomicAdd(&scat[id], c);
    }
    __syncthreads();

    // 5) online max/sum over the 32000-wide vocab row
    const float* frow = fin + (size_t)row * V_;
    float m = -INFINITY, s = 0.0f;
    for (int v = tid; v < V_; v += BLK) {
        float x  = frow[v];
        float nm = fmaxf(m, x);
        s = s * __expf(m - nm) + __expf(x - nm);
        m = nm;
    }
    #pragma unroll
    for (int off = 16; off > 0; off >>= 1) {
        float om = __shfl_xor(m, off);
        float os = __shfl_xor(s, off);
        float nm = fmaxf(m, om);
        s = s * __expf(m - nm) + os * __expf(om - nm);
        m = nm;
    }
    if (lane == 0) { red[2 * wave] = m; red[2 * wave + 1] = s; }
    __syncthreads();
    if (tid < 32) {
        float m2 = red[2 * tid];
        float s2 = red[2 * tid + 1];
        #pragma unroll
        for (int off = 16; off > 0; off >>= 1) {
            float om = __shfl_xor(m2, off);
            float os = __shfl_xor(s2, off);
            float nm = fmaxf(m2, om);
            s2 = s2 * __expf(m2 - nm) + os * __expf(om - nm);
            m2 = nm;
        }
        if (tid == 0) { red[0] = m2; red[1] = s2; }
    }
    __syncthreads();

    const float M2    = red[0];
    const float scale = pg / red[1];

    // 6) final streaming pass: log(pg*softmax + scatter). frow re-read hits L2.
    float* orow = out + (size_t)row * V_;
    for (int v = tid; v < V_; v += BLK) {
        float x = frow[v];
        float val = scale * __expf(x - M2) + scat[v];
        orow[v] = __logf(val);
    }
}

extern "C" void kernel_launch(void* const* d_in, const int* in_sizes, int n_in,
                              void* d_out, int out_size, void* d_ws, size_t ws_size,
                              hipStream_t stream) {
    const float* dec  = (const float*)d_in[0];  // [16,128,768]
    const float* fin  = (const float*)d_in[1];  // [16,128,32000]
    const float* attn = (const float*)d_in[2];  // [16,12,128,512]
    const int*   enc  = (const int*)  d_in[3];  // [16,512]
    const float* Wp   = (const float*)d_in[4];  // [768,1]
    const float* bp   = (const float*)d_in[5];  // [1]
    float* out = (float*)d_out;

    float* pgen = (float*)d_ws;                 // 2048 floats (8 KB)

    // Kernel 1: WMMA f32 GEMV for p_gen. 2048 rows / 16 per wave = 128 blocks.
    pgen_wmma_kernel<<<128, 32, 0, stream>>>(dec, Wp, bp, pgen);

    // Kernel 2: fused row kernel, dynamic LDS = (V + 512 + 64) floats ~ 130 KB.
    size_t ldsBytes = (size_t)(V_ + L_ + 64) * sizeof(float);
    dim3 grid(T_, B_);
    fused_row_kernel<<<grid, BLK, ldsBytes, stream>>>(fin, attn, enc, pgen, out);
}